// GaussianAttention_37735582662867
// MI455X (gfx1250) — compile-verified
//
#include <hip/hip_runtime.h>

// ---------------------------------------------------------------------------
// Gaussian attention window (Graves) for MI455X / gfx1250.
//
// Roofline: onehots stream (335 MB) dominates -> ~14.4 us at 23.3 TB/s HBM.
// GEMM stage (98 MFLOP) done with V_WMMA_F32_16X16X4_F32 (f32 A/B, K=400
// divisible by 4). Einsum stage is a per-batch GEMV (different matrix per b),
// not WMMA-expressible -> pure bandwidth kernel with dense float4 streaming.
// ---------------------------------------------------------------------------

typedef float v2f __attribute__((ext_vector_type(2)));
typedef float v8f __attribute__((ext_vector_type(8)));

#define B_TOT 4096
#define U_LEN 256
#define V_VOC 80
#define K_MIX 10
#define D_HID 400
#define KP    (D_HID / 2)   // 200 f32-pairs along K

// flat output layout (floats): w_t | alpha | beta | kappa | phi
#define OUT_WT    0
#define OUT_ALPHA (B_TOT * V_VOC)            // 327680
#define OUT_BETA  (OUT_ALPHA + B_TOT * K_MIX)
#define OUT_KAPPA (OUT_BETA + B_TOT * K_MIX)
#define OUT_PHI   (OUT_KAPPA + B_TOT * K_MIX)

// Swizzled LDS index for W^T pairs: odd pair-rows rotate columns by 16 so the
// two half-wave ds_load_b64's each hit all 64 banks (no 2-way conflict).
__device__ __forceinline__ int lw_idx(int p, int n) {
  return p * 32 + ((n + ((p & 1) << 4)) & 31);
}

// ---------------------------------------------------------------------------
// Kernel A: mp = exp(h @ W^T + b); alpha/beta/kappa -> d_out regions.
// 1 wave = one 16-batch tile, both N-halves (N=0..31, 30 valid).
// A frag (16x4 f32): lane<16 holds h[b0+lane, d0+0/1]; lane>=16 holds d0+2/3.
// B frag (4x16 f32): lane<16 holds W^T[d0+0/1, n]; lane>=16 holds d0+2/3.
// ---------------------------------------------------------------------------
__global__ __launch_bounds__(256)
void ga_mixture_params_kernel(const float* __restrict__ h,
                              const float* __restrict__ kappa_prev,
                              const float* __restrict__ W,
                              const float* __restrict__ bias,
                              float* __restrict__ out) {
  __shared__ v2f   lw[KP * 32];   // W^T as (K-pair, N) f32 pairs: 51200 B
  __shared__ float lbias[32];

  const int tid = threadIdx.x;

  // Stage W^T into LDS: lw[p][n] = (W[n, 2p], W[n, 2p+1]), zero for n >= 30.
  for (int e = tid; e < KP * 32; e += 256) {
    int p = e >> 5, n = e & 31;
    v2f v;
    if (n < 30) {
      v.x = W[n * D_HID + 2 * p];
      v.y = W[n * D_HID + 2 * p + 1];
    } else {
      v.x = 0.f; v.y = 0.f;
    }
    lw[lw_idx(p, n)] = v;
  }
  if (tid < 32) lbias[tid] = (tid < 30) ? bias[tid] : 0.f;
  __syncthreads();

  const int wave = tid >> 5;
  const int lane = tid & 31;
  const int hi   = lane >> 4;   // 0: K=0,1  1: K=2,3
  const int col  = lane & 15;
  const int b0   = blockIdx.x * 128 + wave * 16;

  // per-lane A row pointer (global_load_b64 per K-step)
  const float* arow = h + (size_t)(b0 + col) * D_HID + 2 * hi;

  v8f c0 = {};  // columns n = 0..15
  v8f c1 = {};  // columns n = 16..31
  for (int d0 = 0; d0 < D_HID; d0 += 4) {
    v2f a  = *(const v2f*)(arow + d0);
    int p  = (d0 >> 1) + hi;
    v2f bl = lw[lw_idx(p, col)];
    v2f bh = lw[lw_idx(p, 16 + col)];
    c0 = __builtin_amdgcn_wmma_f32_16x16x4_f32(false, a, false, bl,
                                               (short)0, c0, false, false);
    c1 = __builtin_amdgcn_wmma_f32_16x16x4_f32(false, a, false, bh,
                                               (short)0, c1, false, false);
  }

  // Epilogue. C/D layout: VGPR i = row (i + 8*hi), column col (+16 for c1).
#pragma unroll
  for (int i = 0; i < 8; ++i) {
    const int m  = i + 8 * hi;
    const int gb = b0 + m;

    // half 0: n = col in 0..15 -> alpha (0..9) / beta (10..15)
    float v0 = __expf(c0[i] + lbias[col]);
    if (col < 10) {
      out[OUT_ALPHA + gb * K_MIX + col] = v0;
    } else {
      out[OUT_BETA + gb * K_MIX + (col - 10)] = v0;
    }

    // half 1: n = 16+col in 16..31 -> beta (16..19) / kappa (20..29)
    const int n1 = 16 + col;
    float v1 = __expf(c1[i] + lbias[n1]);
    if (n1 < 20) {
      out[OUT_BETA + gb * K_MIX + (n1 - 10)] = v1;
    } else if (n1 < 30) {
      const int kk = n1 - 20;
      out[OUT_KAPPA + gb * K_MIX + kk] = v1 + kappa_prev[gb * K_MIX + kk];
    }
  }
}

// ---------------------------------------------------------------------------
// Kernel P: phi[b,u] = mask[b,u] * sum_k alpha*exp(-beta*(kappa-u)^2).
// 8 batches per block; mixture params in an LDS float4 table so the inner
// loop is one ds_load_b128 per mixture; thread t owns column u=t (coalesced
// mask reads / phi stores).
// ---------------------------------------------------------------------------
__global__ __launch_bounds__(256)
void ga_phi_kernel(const float* __restrict__ mask,
                   const float* __restrict__ out_abk,  // out + OUT_ALPHA
                   float* __restrict__ out_phi) {      // out + OUT_PHI
  __shared__ float4 abk[8 * K_MIX];  // [lb][k] = {alpha, beta, kappa, pad}

  const int tid  = threadIdx.x;
  const int base = blockIdx.x * 8 * K_MIX;  // 80 params per region per block

  if (tid < 3 * 8 * K_MIX) {
    const int r = tid / (8 * K_MIX);        // 0=alpha 1=beta 2=kappa
    const int j = tid % (8 * K_MIX);
    float v = out_abk[(size_t)r * B_TOT * K_MIX + base + j];
    ((float*)&abk[j])[r] = v;
  }
  __syncthreads();

  const int   u  = tid;
  const float uf = (float)u;
  const int   b0 = blockIdx.x * 8;

  for (int lb = 0; lb < 8; ++lb) {
    float s = 0.f;
#pragma unroll
    for (int k = 0; k < K_MIX; ++k) {
      float4 p = abk[lb * K_MIX + k];       // ds_load_b128 broadcast
      float  d = p.z - uf;
      s += p.x * __expf(-p.y * d * d);
    }
    const size_t gi = (size_t)(b0 + lb) * U_LEN + u;
    out_phi[gi] = s * mask[gi];
  }
}

// ---------------------------------------------------------------------------
// Kernel B: w_t[b,v] = sum_u phi[b,u] * onehots[b,u,v].
// Bandwidth-bound (335 MB stream). One block per batch, 160 thr = 20 float4
// columns x 8 u-groups; every 320 B row is consumed as dense global_load_b128.
// ---------------------------------------------------------------------------
__global__ __launch_bounds__(160)
void ga_window_kernel(const float* __restrict__ onehots,
                      const float* __restrict__ phi,   // out + OUT_PHI
                      float* __restrict__ wt) {        // out + OUT_WT
  __shared__ float  lphi[U_LEN];
  __shared__ float4 red[160];

  const int b = blockIdx.x;
  const int t = threadIdx.x;

  for (int j = t; j < U_LEN; j += 160)
    lphi[j] = phi[(size_t)b * U_LEN + j];
  __syncthreads();

  const int vq = t % 20;           // float4 column (v = 4*vq .. 4*vq+3)
  const int g  = t / 20;           // u-group, 32 rows each
  const float4* oh = (const float4*)(onehots + (size_t)b * U_LEN * V_VOC);

  float4 acc = make_float4(0.f, 0.f, 0.f, 0.f);
  const int u0 = g * 32;
#pragma unroll 4
  for (int i = 0; i < 32; ++i) {
    const int u = u0 + i;
    const float  p = lphi[u];
    const float4 o = oh[u * 20 + vq];      // global_load_b128, dense rows
    acc.x += p * o.x; acc.y += p * o.y;
    acc.z += p * o.z; acc.w += p * o.w;
  }

  red[t] = acc;
  __syncthreads();
  if (t < 20) {
    float4 s = red[t];
#pragma unroll
    for (int j = 1; j < 8; ++j) {
      float4 r = red[t + 20 * j];
      s.x += r.x; s.y += r.y; s.z += r.z; s.w += r.w;
    }
    ((float4*)(wt + (size_t)b * V_VOC))[t] = s;  // global_store_b128
  }
}

// ---------------------------------------------------------------------------
extern "C" void kernel_launch(void* const* d_in, const int* in_sizes, int n_in,
                              void* d_out, int out_size, void* d_ws,
                              size_t ws_size, hipStream_t stream) {
  (void)in_sizes; (void)n_in; (void)out_size; (void)d_ws; (void)ws_size;

  const float* h_t   = (const float*)d_in[0];  // [4096,400]
  const float* kprev = (const float*)d_in[1];  // [4096,10]
  const float* mask  = (const float*)d_in[2];  // [4096,256]
  const float* oneh  = (const float*)d_in[3];  // [4096,256,80]
  const float* W     = (const float*)d_in[4];  // [30,400]
  const float* bias  = (const float*)d_in[5];  // [30]
  float* out = (float*)d_out;

  ga_mixture_params_kernel<<<B_TOT / 128, 256, 0, stream>>>(
      h_t, kprev, W, bias, out);
  ga_phi_kernel<<<B_TOT / 8, 256, 0, stream>>>(
      mask, out + OUT_ALPHA, out + OUT_PHI);
  ga_window_kernel<<<B_TOT, 160, 0, stream>>>(
      oneh, out + OUT_PHI, out + OUT_WT);
}